// GNNActor_8091718386001
// MI455X (gfx1250) — compile-verified
//
#include <hip/hip_runtime.h>
#include <hip/hip_bf16.h>

#define N_NODES 60000
#define N_EDGES 960000
#define C_IN    64
#define H_DIM   32

typedef __attribute__((ext_vector_type(2))) float v2f;
typedef __attribute__((ext_vector_type(8))) float v8f;

// gfx1250 has native global_atomic_add_f32; unsafeAtomicAdd guarantees the HW path
#define FATOMIC(p, v) unsafeAtomicAdd((p), (v))

// ---------------------------------------------------------------------------
// deg[i] = 1 (self loop), global concentration sum = 0
__global__ __launch_bounds__(256) void init_kernel(float* __restrict__ deg,
                                                   float* __restrict__ gsum) {
    int i = blockIdx.x * blockDim.x + threadIdx.x;
    if (i < N_NODES) deg[i] = 1.0f;
    if (i == 0) *gsum = 0.0f;
}

// deg[col[e]] += 1
__global__ __launch_bounds__(256) void degree_kernel(const int* __restrict__ col,
                                                     float* __restrict__ deg) {
    int e = blockIdx.x * blockDim.x + threadIdx.x;
    if (e < N_EDGES) FATOMIC(&deg[col[e]], 1.0f);
}

// deg -> 1/sqrt(deg) in place (deg >= 1 always due to self loops)
__global__ __launch_bounds__(256) void dinv_kernel(float* __restrict__ deg) {
    int i = blockIdx.x * blockDim.x + threadIdx.x;
    if (i < N_NODES) deg[i] = rsqrtf(deg[i]);
}

// ---------------------------------------------------------------------------
// xw = state @ W_gcn  via V_WMMA_F32_16X16X4_F32.
// One wave per 16-row tile; 16 K-chunks x 4 column blocks = 64 WMMA / wave.
// W is staged TRANSPOSED in LDS with a padded stride of 66 dwords so every
// B fragment {W[k][n], W[k+1][n]} is one 8B-aligned ds_load_b64 into an
// even-aligned VGPR pair (WMMA operand requirement), with no re-pairing movs.
#define WT_STRIDE 66   // even (b64 alignment since k is even) + bank skew

__global__ __launch_bounds__(256) void xw_wmma_kernel(const float* __restrict__ state,
                                                      const float* __restrict__ Wg,
                                                      float* __restrict__ xw) {
    __shared__ float sWt[C_IN * WT_STRIDE];   // 16.5 KB
    for (int i = threadIdx.x; i < C_IN * C_IN; i += blockDim.x) {
        const int k = i >> 6;        // row of W (K index)
        const int n = i & 63;        // col of W (N index)
        sWt[n * WT_STRIDE + k] = Wg[i];   // bank = (2n+k)&63 -> conflict-free
    }
    __syncthreads();

    const int wavesPerBlock = blockDim.x >> 5;
    const int tile = blockIdx.x * wavesPerBlock + (threadIdx.x >> 5);
    if (tile >= N_NODES / 16) return;

    const int lane = threadIdx.x & 31;
    const int half = lane >> 4;   // K-pair selector for A/B fragments
    const int mrow = lane & 15;   // M (A) / N (B,D) index within tile

    const float* __restrict__ arow = state + (size_t)(tile * 16 + mrow) * C_IN;
    const float* __restrict__ bc0 = sWt + (0 * 16 + mrow) * WT_STRIDE;
    const float* __restrict__ bc1 = sWt + (1 * 16 + mrow) * WT_STRIDE;
    const float* __restrict__ bc2 = sWt + (2 * 16 + mrow) * WT_STRIDE;
    const float* __restrict__ bc3 = sWt + (3 * 16 + mrow) * WT_STRIDE;

    v8f acc0 = {}, acc1 = {}, acc2 = {}, acc3 = {};

#pragma unroll
    for (int kc = 0; kc < 16; ++kc) {
        const int k = kc * 4 + half * 2;            // this lane-half's K pair
        // A fragment (16x4 f32): lane<16 -> K {0,1}, lane>=16 -> K {2,3}
        const v2f a = *(const v2f*)(arow + k);      // 8B aligned (k even)
        // B fragments (4x16 f32): contiguous transposed pairs, one b64 each
        const v2f b0 = *(const v2f*)(bc0 + k);
        const v2f b1 = *(const v2f*)(bc1 + k);
        const v2f b2 = *(const v2f*)(bc2 + k);
        const v2f b3 = *(const v2f*)(bc3 + k);

        acc0 = __builtin_amdgcn_wmma_f32_16x16x4_f32(false, a, false, b0, (short)0, acc0, false, false);
        acc1 = __builtin_amdgcn_wmma_f32_16x16x4_f32(false, a, false, b1, (short)0, acc1, false, false);
        acc2 = __builtin_amdgcn_wmma_f32_16x16x4_f32(false, a, false, b2, (short)0, acc2, false, false);
        acc3 = __builtin_amdgcn_wmma_f32_16x16x4_f32(false, a, false, b3, (short)0, acc3, false, false);
    }

    // D layout: VGPR r -> M = r + 8*half, N = mrow (+ 16*nb)
    float* __restrict__ orow = xw + (size_t)(tile * 16 + half * 8) * C_IN + mrow;
#pragma unroll
    for (int r = 0; r < 8; ++r) {
        orow[r * C_IN + 0]  = acc0[r];
        orow[r * C_IN + 16] = acc1[r];
        orow[r * C_IN + 32] = acc2[r];
        orow[r * C_IN + 48] = acc3[r];
    }
}

// ---------------------------------------------------------------------------
// agg[col] += xw[row] * dinv[row]*dinv[col]   (one wave32 per edge, 2 cols/lane)
__global__ __launch_bounds__(256) void scatter_kernel(const int* __restrict__ eidx,
                                                      const float* __restrict__ dinv,
                                                      const float* __restrict__ xw,
                                                      float* __restrict__ agg) {
    const int* __restrict__ rowI = eidx;
    const int* __restrict__ colI = eidx + N_EDGES;
    const int lane   = threadIdx.x & 31;
    const int wave   = (blockIdx.x * blockDim.x + threadIdx.x) >> 5;
    const int nwaves = (gridDim.x * blockDim.x) >> 5;
    for (int e = wave; e < N_EDGES; e += nwaves) {
        const int r = rowI[e];
        const int c = colI[e];
        const float nrm = dinv[r] * dinv[c];
        const float2 v = *(const float2*)(xw + (size_t)r * C_IN + lane * 2);
        float* dst = agg + (size_t)c * C_IN + lane * 2;
        FATOMIC(dst,     v.x * nrm);   // non-returning -> global_atomic_add_f32
        FATOMIC(dst + 1, v.y * nrm);
    }
}

// ---------------------------------------------------------------------------
// Fused: self-loop term + b_gcn + ReLU + residual, then 64->32->32->1 MLP,
// softplus, per-block sum -> one atomic. One wave32 per node.
__global__ __launch_bounds__(256) void mlp_kernel(const float* __restrict__ state,
                                                  const float* __restrict__ xw,
                                                  const float* __restrict__ agg,
                                                  const float* __restrict__ dinv,
                                                  const float* __restrict__ b_gcn,
                                                  const float* __restrict__ W1,
                                                  const float* __restrict__ b1,
                                                  const float* __restrict__ W2,
                                                  const float* __restrict__ b2,
                                                  const float* __restrict__ W3,
                                                  const float* __restrict__ b3,
                                                  float* __restrict__ conc,
                                                  float* __restrict__ gsum) {
    __shared__ float sW1[C_IN * H_DIM];
    __shared__ float sW2[H_DIM * H_DIM];
    __shared__ float sW3[H_DIM], sB1[H_DIM], sB2[H_DIM];
    __shared__ float sPart[8];

    for (int i = threadIdx.x; i < C_IN * H_DIM; i += blockDim.x) sW1[i] = W1[i];
    for (int i = threadIdx.x; i < H_DIM * H_DIM; i += blockDim.x) sW2[i] = W2[i];
    if (threadIdx.x < H_DIM) {
        sW3[threadIdx.x] = W3[threadIdx.x];
        sB1[threadIdx.x] = b1[threadIdx.x];
        sB2[threadIdx.x] = b2[threadIdx.x];
    }
    __syncthreads();

    const int lane   = threadIdx.x & 31;
    const int wid    = threadIdx.x >> 5;
    const int wave   = (blockIdx.x * blockDim.x + threadIdx.x) >> 5;
    const int nwaves = (gridDim.x * blockDim.x) >> 5;

    const float b3v = b3[0];
    const float bg0 = b_gcn[lane * 2];
    const float bg1 = b_gcn[lane * 2 + 1];

    float wsum = 0.0f;
    for (int n = wave; n < N_NODES; n += nwaves) {
        const float dn  = dinv[n];
        const float dn2 = dn * dn;
        const size_t base = (size_t)n * C_IN + lane * 2;
        const float2 ag = *(const float2*)(agg + base);
        const float2 xv = *(const float2*)(xw + base);
        const float2 st = *(const float2*)(state + base);
        // agg + self-loop message + bias, ReLU, residual
        const float x0 = fmaxf(ag.x + xv.x * dn2 + bg0, 0.0f) + st.x;
        const float x1 = fmaxf(ag.y + xv.y * dn2 + bg1, 0.0f) + st.y;

        // layer 1: h1[lane] = sum_k x[k] * W1[k][lane]
        float h1 = 0.0f;
#pragma unroll
        for (int j = 0; j < 32; ++j) {
            const float xa = __shfl(x0, j, 32);      // k = 2j
            const float xb = __shfl(x1, j, 32);      // k = 2j+1
            h1 += xa * sW1[(2 * j) * H_DIM + lane] + xb * sW1[(2 * j + 1) * H_DIM + lane];
        }
        h1 += sB1[lane];
        h1 = h1 > 0.0f ? h1 : 0.01f * h1;            // leaky_relu(0.01)

        // layer 2
        float h2 = 0.0f;
#pragma unroll
        for (int j = 0; j < 32; ++j) {
            h2 += __shfl(h1, j, 32) * sW2[j * H_DIM + lane];
        }
        h2 += sB2[lane];
        h2 = h2 > 0.0f ? h2 : 0.01f * h2;

        // layer 3: scalar per node via wave reduction
        float t = h2 * sW3[lane];
#pragma unroll
        for (int o = 16; o > 0; o >>= 1) t += __shfl_xor(t, o, 32);
        const float z  = t + b3v;
        const float cv = (z > 20.0f) ? z : log1pf(__expf(z));  // stable softplus

        if (lane == 0) { conc[n] = cv; wsum += cv; }
    }

    if (lane == 0) sPart[wid] = wsum;
    __syncthreads();
    if (threadIdx.x == 0) {
        float s = 0.0f;
        const int nw = blockDim.x >> 5;
        for (int w = 0; w < nw; ++w) s += sPart[w];
        FATOMIC(gsum, s);
    }
}

// ---------------------------------------------------------------------------
__global__ __launch_bounds__(256) void normalize_kernel(const float* __restrict__ conc,
                                                        const float* __restrict__ gsum,
                                                        float* __restrict__ out) {
    int i = blockIdx.x * blockDim.x + threadIdx.x;
    if (i < N_NODES) out[i] = conc[i] / (*gsum + 1e-20f);
}

// ---------------------------------------------------------------------------
extern "C" void kernel_launch(void* const* d_in, const int* in_sizes, int n_in,
                              void* d_out, int out_size, void* d_ws, size_t ws_size,
                              hipStream_t stream) {
    const float* state = (const float*)d_in[0];
    const float* Wg    = (const float*)d_in[1];
    const float* bg    = (const float*)d_in[2];
    const float* W1    = (const float*)d_in[3];
    const float* b1    = (const float*)d_in[4];
    const float* W2    = (const float*)d_in[5];
    const float* b2    = (const float*)d_in[6];
    const float* W3    = (const float*)d_in[7];
    const float* b3    = (const float*)d_in[8];
    const int*   eidx  = (const int*)d_in[9];
    float*       out   = (float*)d_out;

    // workspace layout (all re-initialized every call)
    float* xw   = (float*)d_ws;                          // [N, C]
    float* agg  = xw   + (size_t)N_NODES * C_IN;         // [N, C]
    float* dinv = agg  + (size_t)N_NODES * C_IN;         // [N] (deg then rsqrt)
    float* conc = dinv + N_NODES;                        // [N]
    float* gsum = conc + N_NODES;                        // [1]

    hipMemsetAsync(agg, 0, (size_t)N_NODES * C_IN * sizeof(float), stream);
    init_kernel<<<(N_NODES + 255) / 256, 256, 0, stream>>>(dinv, gsum);
    degree_kernel<<<(N_EDGES + 255) / 256, 256, 0, stream>>>(eidx + N_EDGES, dinv);
    dinv_kernel<<<(N_NODES + 255) / 256, 256, 0, stream>>>(dinv);
    xw_wmma_kernel<<<(N_NODES / 16 + 7) / 8, 256, 0, stream>>>(state, Wg, xw);
    scatter_kernel<<<4096, 256, 0, stream>>>(eidx, dinv, xw, agg);
    mlp_kernel<<<480, 256, 0, stream>>>(state, xw, agg, dinv, bg,
                                        W1, b1, W2, b2, W3, b3, conc, gsum);
    normalize_kernel<<<(N_NODES + 255) / 256, 256, 0, stream>>>(conc, gsum, out);
}